// GraphAttentionNet_39264591020594
// MI455X (gfx1250) — compile-verified
//
#include <hip/hip_runtime.h>
#include <hip/hip_bf16.h>
#include <math.h>

// Problem constants from the reference
#define B_SZ   4096
#define F_SZ   128
#define KF1_SZ 512
#define N_SZ   64
#define OUT_SZ 8

typedef __attribute__((ext_vector_type(2))) float v2f;
typedef __attribute__((ext_vector_type(8))) float v8f;

// ---------------------------------------------------------------------------
// Kernel 1: w_au[f] = sum_d W[d,f]*a_u[d],  w_av[f] = sum_d W[d,f]*a_v[d]
// W viewed flat as (KF1, F). One block, 128 threads; column reads are
// coalesced across threads, a_w[d] is a broadcast load.
// ---------------------------------------------------------------------------
__global__ void prep_w(const float* __restrict__ W,
                       const float* __restrict__ a_w,
                       float* __restrict__ w_au,
                       float* __restrict__ w_av) {
  int f = threadIdx.x;  // 0..127
  float au = 0.f, av = 0.f;
  for (int d = 0; d < KF1_SZ; ++d) {
    float w = W[d * F_SZ + f];
    au += w * a_w[d];
    av += w * a_w[KF1_SZ + d];
  }
  w_au[f] = au;
  w_av[f] = av;
}

// ---------------------------------------------------------------------------
// Kernel 2: per-b attention. Streams input_n_u exactly once (the 128 MB /
// 23.3 TB/s roofline term). One block per b, 128 threads (4 waves).
//   s_u    = input_u[b] . w_au
//   s_v[n] = X_b[:,n] . w_av
//   alpha  = softmax(leaky_relu(s_u + s_v + a_b))
//   y[f]   = sum_n X_b[f,n] * alpha[n]
// X in LDS with row stride 68 (pad 4): float4 stores stay 16B-aligned,
// column reads (s_v) conflict-free, row reads (y) at most 2-way.
// ---------------------------------------------------------------------------
#define XST 68

__global__ void attn(const float* __restrict__ input_u,
                     const float* __restrict__ input_n_u,
                     const float* __restrict__ w_au,
                     const float* __restrict__ w_av,
                     const float* __restrict__ a_b,
                     float* __restrict__ y) {
  __shared__ float X[F_SZ * XST];
  __shared__ float wauL[F_SZ];
  __shared__ float wavL[F_SZ];
  __shared__ float red[F_SZ];
  __shared__ float sc[N_SZ];
  __shared__ float alpha[N_SZ];

  const int b = blockIdx.x;
  const int t = threadIdx.x;  // 0..127

  wauL[t] = w_au[t];
  wavL[t] = w_av[t];

  // Stage X_b (128 x 64 f32 = 32 KB) via float4: 2048 vec4 / 128 threads.
  const float4* Xg = (const float4*)(input_n_u + (size_t)b * (F_SZ * N_SZ));
  for (int i = 0; i < 16; ++i) {
    int idx4 = t + i * 128;
    float4 v = Xg[idx4];
    int base = idx4 * 4;
    int f = base >> 6;   // row
    int n = base & 63;   // col (multiple of 4)
    *(float4*)&X[f * XST + n] = v;
  }

  // s_u partial products
  red[t] = input_u[(size_t)b * F_SZ + t] * wauL[t];
  __syncthreads();
  for (int s = 64; s > 0; s >>= 1) {
    if (t < s) red[t] += red[t + s];
    __syncthreads();
  }
  const float s_u = red[0];
  const float bias = a_b[0];
  __syncthreads();

  // s_v[n], leaky_relu
  if (t < N_SZ) {
    float acc = 0.f;
    for (int f = 0; f < F_SZ; ++f) acc += X[f * XST + t] * wavL[f];
    float s = s_u + acc + bias;
    sc[t] = (s >= 0.f) ? s : 0.1f * s;
    red[t] = sc[t];
  }
  __syncthreads();
  // max over 64
  for (int s = 32; s > 0; s >>= 1) {
    if (t < s) red[t] = fmaxf(red[t], red[t + s]);
    __syncthreads();
  }
  const float mx = red[0];
  __syncthreads();
  float e = 0.f;
  if (t < N_SZ) {
    e = expf(sc[t] - mx);
    red[t] = e;
  }
  __syncthreads();
  for (int s = 32; s > 0; s >>= 1) {
    if (t < s) red[t] += red[t + s];
    __syncthreads();
  }
  const float denom = red[0];
  __syncthreads();
  if (t < N_SZ) alpha[t] = e / denom;
  __syncthreads();

  // y[f] = sum_n X[f,n] * alpha[n]
  float acc = 0.f;
  const float* Xrow = &X[t * XST];
  for (int n = 0; n < N_SZ; ++n) acc += Xrow[n] * alpha[n];
  y[(size_t)b * F_SZ + t] = acc;
}

// ---------------------------------------------------------------------------
// Kernel 3: agg = Y (4096x128) @ W_flat^T (128x512), fused celu, then the
// 512->8 sigmoid projection. 256 blocks of 256 threads (8 wave32s).
// Each block owns 16 B-rows; each wave owns a 16x64 column strip and runs a
// K=128 loop of v_wmma_f32_16x16x4_f32 (full f32 precision on the matrix
// pipe). A staged in LDS (stride 132 -> conflict-free for the 16x4 f32 A
// layout), H staged in LDS (stride 516 -> conflict-free writes and reads).
//
// f32 16x16x4 WMMA layouts (ISA 7.12.2):
//   A (16x4):  lane l holds M=l&15, K0=(l>>4)*2; vgpr v -> K0+v
//   B (4x16):  lane l holds N=l&15, K0=(l>>4)*2; vgpr v -> K0+v   (mirror)
//   C/D:       lane l holds N=l&15; vgpr v -> M = v + 8*(l>>4)
// ---------------------------------------------------------------------------
#define APAD 132
#define HST  516

__global__ void gemm_out(const float* __restrict__ y,
                         const float* __restrict__ W,
                         const float* __restrict__ h2o_w,
                         const float* __restrict__ h2o_b,
                         float* __restrict__ out) {
  __shared__ float A[16 * APAD];
  __shared__ float H[16 * HST];

  const int t = threadIdx.x;   // 0..255
  const int lane = t & 31;
  const int wave = t >> 5;     // 0..7
  const int b0 = blockIdx.x * 16;

  // Stage A: 16 rows x 128 K of Y
  for (int i = 0; i < 8; ++i) {
    int idx = t + i * 256;     // 0..2047
    int m = idx >> 7;
    int k = idx & 127;
    A[m * APAD + k] = y[(size_t)(b0 + m) * F_SZ + k];
  }
  __syncthreads();

  const int n_base = wave * 64;       // this wave's 64 columns of KF1
  const int mA = lane & 15;
  const int kOff = (lane >> 4) * 2;   // 0 for lanes 0-15, 2 for lanes 16-31

  v8f c[4];
  for (int j = 0; j < 4; ++j)
    for (int vv = 0; vv < 8; ++vv) c[j][vv] = 0.f;

  for (int kk = 0; kk < F_SZ; kk += 4) {
    v2f a;
    a.x = A[mA * APAD + kk + kOff];
    a.y = A[mA * APAD + kk + kOff + 1];
    for (int j = 0; j < 4; ++j) {
      int n = n_base + j * 16 + (lane & 15);
      // B[k][n] = W_flat^T[k][n] = W[n*128 + k]  (W is L2-resident, 256 KB)
      v2f bf;
      bf.x = W[(size_t)n * F_SZ + kk + kOff];
      bf.y = W[(size_t)n * F_SZ + kk + kOff + 1];
      c[j] = __builtin_amdgcn_wmma_f32_16x16x4_f32(
          /*neg_a=*/false, a, /*neg_b=*/false, bf,
          /*c_mod=*/(short)0, c[j], /*reuse_a=*/false, /*reuse_b=*/false);
    }
  }

  // celu -> H in LDS
  for (int j = 0; j < 4; ++j) {
    int n = n_base + j * 16 + (lane & 15);
    for (int vv = 0; vv < 8; ++vv) {
      int m = vv + ((lane >> 4) ? 8 : 0);
      float x = c[j][vv];
      H[m * HST + n] = (x >= 0.f) ? x : expm1f(x);
    }
  }
  __syncthreads();

  // out[b, o] = sigmoid(H[b,:] . h2o_w[o,:] + h2o_b[o]); 16*8 = 128 pairs
  if (t < 128) {
    int bl = t >> 3;
    int o = t & 7;
    float acc = h2o_b[o];
    const float* wrow = h2o_w + (size_t)o * KF1_SZ;
    const float* hrow = &H[bl * HST];
    for (int d = 0; d < KF1_SZ; ++d) acc += hrow[d] * wrow[d];
    out[(size_t)(b0 + bl) * OUT_SZ + o] = 1.f / (1.f + expf(-acc));
  }
}

// ---------------------------------------------------------------------------
// Launch
// ---------------------------------------------------------------------------
extern "C" void kernel_launch(void* const* d_in, const int* in_sizes, int n_in,
                              void* d_out, int out_size, void* d_ws, size_t ws_size,
                              hipStream_t stream) {
  const float* input_u   = (const float*)d_in[0];  // (B, F)
  const float* input_n_u = (const float*)d_in[1];  // (B, F, N)
  const float* W         = (const float*)d_in[2];  // (K, F1, F) = (512,128) flat
  const float* a_w       = (const float*)d_in[3];  // (1, 1024)
  const float* a_b       = (const float*)d_in[4];  // (1,)
  const float* h2o_w     = (const float*)d_in[5];  // (8, 512)
  const float* h2o_b     = (const float*)d_in[6];  // (8,)
  float* out = (float*)d_out;                      // (B, 8)

  // Workspace layout: w_au[128] | w_av[128] | y[B*128]  (~2.1 MB)
  float* ws   = (float*)d_ws;
  float* w_au = ws;
  float* w_av = ws + F_SZ;
  float* yb   = ws + 2 * F_SZ;

  prep_w<<<1, F_SZ, 0, stream>>>(W, a_w, w_au, w_av);
  attn<<<B_SZ, F_SZ, 0, stream>>>(input_u, input_n_u, w_au, w_av, a_b, yb);
  gemm_out<<<B_SZ / 16, 256, 0, stream>>>(yb, W, h2o_w, h2o_b, out);
}